// YOLOv3Postprocessor_80418967650418
// MI455X (gfx1250) — compile-verified
//
#include <hip/hip_runtime.h>
#include <stdint.h>

// ---------------- constants from the reference ----------------
#define NANCH     25200     // 3*(80^2+40^2+20^2) anchors per image
#define TOPK      2048
#define MAXDET    300
#define BATCH     16
#define SORT_N    32768     // next pow2 >= NANCH
#define CONF_T    0.001f
#define IOU_T     0.45f
#define CLS_OFF   1.0e4f

__constant__ float c_anchors[9][2] = {
    {10.f,13.f},{16.f,30.f},{33.f,23.f},
    {30.f,61.f},{62.f,45.f},{59.f,119.f},
    {116.f,90.f},{156.f,198.f},{373.f,326.f}};

// ---------------- gfx1250 feature probes ----------------
#if defined(__has_builtin)
#  if __has_builtin(__builtin_amdgcn_global_load_async_to_lds_b128)
#    define HAVE_ASYNC_LDS 1
#  endif
#  if __has_builtin(__builtin_amdgcn_s_wait_asynccnt)
#    define HAVE_WAIT_ASYNC 1
#  endif
#endif

#if defined(HAVE_ASYNC_LDS)
typedef __attribute__((ext_vector_type(4))) int v4i;
typedef __attribute__((address_space(1))) v4i* gv4i_p;   // global b128 chunks
typedef __attribute__((address_space(3))) v4i* lv4i_p;   // LDS b128 chunks
#endif

__device__ __forceinline__ void wait_async0() {
#if defined(HAVE_WAIT_ASYNC)
    __builtin_amdgcn_s_wait_asynccnt(0);
#elif defined(HAVE_ASYNC_LDS)
    asm volatile("s_wait_asynccnt 0x0" ::: "memory");
#endif
}

__device__ __forceinline__ float sigmoidf_(float x) {
    return 1.0f / (1.0f + __expf(-x));
}

// =====================================================================
// Kernel 1: decode.  One lane per anchor; block stages 256 contiguous
// 85-float anchor records (87040 B) into LDS via async global->LDS copy.
// =====================================================================
extern "C" __global__ void __launch_bounds__(256)
yolo_decode(const float* __restrict__ in,
            float* __restrict__ boxes_w,      // [B, NANCH, 4]
            float* __restrict__ scores_w,     // [B, NANCH]
            int*   __restrict__ labels_w,     // [B, NANCH]
            int S, int stride, int levelBase, int nOffset,
            const int* __restrict__ imgh, const int* __restrict__ imgw)
{
    extern __shared__ float lds[];            // 256*85 floats = 87040 B
    const int tid = threadIdx.x;
    const long long anchor0 = (long long)blockIdx.x * 256;
    const long long gbase   = anchor0 * 85;   // float offset; 16B aligned per block

#if defined(HAVE_ASYNC_LDS)
    {
        gv4i_p gsrc = (gv4i_p)(in + gbase);
        lv4i_p ldst = (lv4i_p)lds;
        for (int i = tid; i < 5440; i += 256)   // 5440 b128 chunks = 256*85 floats
            __builtin_amdgcn_global_load_async_to_lds_b128(gsrc + i, ldst + i, 0, 0);
        wait_async0();
    }
#else
    {
        const float4* g4 = (const float4*)(in + gbase);
        float4* l4 = (float4*)lds;
        for (int i = tid; i < 5440; i += 256) l4[i] = g4[i];
    }
#endif
    __syncthreads();

    const float* p = lds + tid * 85;
    const float tx = p[0], ty = p[1], tw = p[2], th_ = p[3], tobj = p[4];

    // argmax of logits == argmax of softmax; max softmax prob = 1/sum(exp(l-m))
    float m = p[5]; int arg = 0;
    #pragma unroll 4
    for (int c = 1; c < 80; ++c) {
        float v = p[5 + c];
        if (v > m) { m = v; arg = c; }
    }
    float se = 0.f;
    #pragma unroll 4
    for (int c = 0; c < 80; ++c) se += __expf(p[5 + c] - m);
    const float score = sigmoidf_(tobj) / se;

    const long long gid = anchor0 + tid;      // [b, y, x, a], a fastest
    const int a   = (int)(gid % 3);
    const long long cell = gid / 3;
    const int SS  = S * S;
    const int b   = (int)(cell / SS);
    const int rem = (int)(cell % SS);
    const int gy  = rem / S, gx = rem % S;

    const float cx = (sigmoidf_(tx) + (float)gx) * (float)stride;
    const float cy = (sigmoidf_(ty) + (float)gy) * (float)stride;
    const float bw = __expf(fminf(tw, 8.f));
    const float bh = __expf(fminf(th_, 8.f));
    // anchor_cell*stride == anchors in pixels
    const float pw = c_anchors[levelBase + a][0] * bw;
    const float ph = c_anchors[levelBase + a][1] * bh;
    const float Wm1 = (float)(imgw[0] - 1);
    const float Hm1 = (float)(imgh[0] - 1);
    const float x1 = fminf(fmaxf(cx - 0.5f * pw, 0.f), Wm1);
    const float y1 = fminf(fmaxf(cy - 0.5f * ph, 0.f), Hm1);
    const float x2 = fminf(fmaxf(cx + 0.5f * pw, 0.f), Wm1);
    const float y2 = fminf(fmaxf(cy + 0.5f * ph, 0.f), Hm1);

    const long long n = (long long)b * NANCH + nOffset + (long long)rem * 3 + a;
    ((float4*)boxes_w)[n] = make_float4(x1, y1, x2, y2);
    scores_w[n] = score;
    labels_w[n] = arg;
}

// =====================================================================
// Kernel 2: per-image top-2048 (in-LDS bitonic over 32768 keys, 256 KB
// LDS -- only possible because CDNA5 has 320 KB/WGP), then greedy NMS
// with class-offset boxes, then padded 300-slot output.
// =====================================================================
extern "C" __global__ void __launch_bounds__(1024)
yolo_nms(const float* __restrict__ boxes_w,
         const float* __restrict__ scores_w,
         const int*   __restrict__ labels_w,
         float* __restrict__ out)
{
    extern __shared__ char smem[];
    unsigned long long* keys = (unsigned long long*)smem;        // [32768] 262144 B
    // aux arrays alias keys[2048..] (unused after the sort's top slice)
    float4* offb = (float4*)(smem + 16384);                      // [2048]  32 KB
    int*    keep = (int*)   (smem + 49152);                      // [2048]   8 KB
    int*    rnk  = (int*)   (smem + 57344);                      // [2048]   8 KB

    const int tid = threadIdx.x;
    const int b   = blockIdx.x;
    const float*  sc_g = scores_w + (size_t)b * NANCH;
    const float4* bx_g = (const float4*)boxes_w + (size_t)b * NANCH;
    const int*    lb_g = labels_w + (size_t)b * NANCH;

    // ---- build sort keys: (score_bits << 32) | ~idx  (scores are >0) ----
    for (int i = tid; i < SORT_N; i += 1024) {
        unsigned long long k = 0ull;
        if (i < NANCH) {
            __builtin_prefetch(sc_g + i + 1024, 0, 0);   // global_prefetch_b8
            const float s = sc_g[i];
            k = ((unsigned long long)__float_as_uint(s) << 32)
              | (unsigned)(0xFFFFFFFFu - (unsigned)i);   // smaller idx wins ties
        }
        keys[i] = k;
    }
    __syncthreads();

    // ---- bitonic sort, descending ----
    for (int k = 2; k <= SORT_N; k <<= 1) {
        for (int j = k >> 1; j >= 1; j >>= 1) {
            for (int i = tid; i < SORT_N; i += 1024) {
                const int ixj = i ^ j;
                if (ixj > i) {
                    const unsigned long long a = keys[i];
                    const unsigned long long c = keys[ixj];
                    const bool desc = ((i & k) == 0);
                    if (desc ? (a < c) : (a > c)) { keys[i] = c; keys[ixj] = a; }
                }
            }
            __syncthreads();
        }
    }

    // ---- top-2048 candidates: class-offset boxes + validity ----
    for (int i = tid; i < TOPK; i += 1024) {
        const unsigned long long k = keys[i];
        const float    s   = __uint_as_float((unsigned)(k >> 32));
        const unsigned idx = 0xFFFFFFFFu - (unsigned)(k & 0xFFFFFFFFull);
        const int valid = (s > CONF_T) && (idx < NANCH);
        float4 bx = make_float4(0.f, 0.f, 0.f, 0.f);
        int lbl = 0;
        if (valid) { lbl = lb_g[idx]; bx = bx_g[idx]; }
        const float off = (float)lbl * CLS_OFF;
        offb[i] = make_float4(bx.x + off, bx.y + off, bx.z + off, bx.w + off);
        keep[i] = valid;
    }
    __syncthreads();

    // ---- greedy suppression scan (== reference lax.scan over IoU rows) ----
    for (int i = 0; i < TOPK; ++i) {
        if (keep[i]) {
            const float4 bi = offb[i];
            const float  ai = (bi.z - bi.x) * (bi.w - bi.y);
            for (int j = i + 1 + tid; j < TOPK; j += 1024) {
                if (keep[j]) {
                    const float4 bj = offb[j];
                    const float xx1 = fmaxf(bi.x, bj.x), yy1 = fmaxf(bi.y, bj.y);
                    const float xx2 = fminf(bi.z, bj.z), yy2 = fminf(bi.w, bj.w);
                    const float w = fmaxf(xx2 - xx1, 0.f);
                    const float h = fmaxf(yy2 - yy1, 0.f);
                    const float inter = w * h;
                    const float aj = (bj.z - bj.x) * (bj.w - bj.y);
                    const float iou = inter / (ai + aj - inter + 1e-9f);
                    if (iou > IOU_T) keep[j] = 0;
                }
            }
        }
        __syncthreads();
    }

    // ---- ranks of kept entries (kept entries are already score-descending) ----
    if (tid == 0) {
        int c = 0;
        for (int i = 0; i < TOPK; ++i) { rnk[i] = c; c += keep[i]; }
    }
    __syncthreads();

    // ---- output: [B,300,4] boxes | [B,300] scores | [B,300] labels | [B,300] valid
    float* ob = out;
    float* os = out + (size_t)BATCH * MAXDET * 4;
    float* ol = os  + (size_t)BATCH * MAXDET;
    float* ov = ol  + (size_t)BATCH * MAXDET;

    for (int i = tid; i < MAXDET; i += 1024) {
        ((float4*)ob)[(size_t)b * MAXDET + i] = make_float4(0.f, 0.f, 0.f, 0.f);
        os[(size_t)b * MAXDET + i] = 0.f;
        ol[(size_t)b * MAXDET + i] = 0.f;
        ov[(size_t)b * MAXDET + i] = 0.f;
    }
    __syncthreads();

    for (int i = tid; i < TOPK; i += 1024) {
        if (keep[i] && rnk[i] < MAXDET) {
            const int slot = rnk[i];
            const unsigned long long k = keys[i];
            const float    s   = __uint_as_float((unsigned)(k >> 32));
            const unsigned idx = 0xFFFFFFFFu - (unsigned)(k & 0xFFFFFFFFull);
            ((float4*)ob)[(size_t)b * MAXDET + slot] = bx_g[idx];
            os[(size_t)b * MAXDET + slot] = s;
            ol[(size_t)b * MAXDET + slot] = (float)lb_g[idx];
            ov[(size_t)b * MAXDET + slot] = 1.f;
        }
    }
}

// =====================================================================
extern "C" void kernel_launch(void* const* d_in, const int* in_sizes, int n_in,
                              void* d_out, int out_size, void* d_ws, size_t ws_size,
                              hipStream_t stream) {
    (void)in_sizes; (void)n_in; (void)out_size; (void)ws_size;
    const float* p0  = (const float*)d_in[0];   // [16,80,80,255]
    const float* p1  = (const float*)d_in[1];   // [16,40,40,255]
    const float* p2  = (const float*)d_in[2];   // [16,20,20,255]
    const int* imgh  = (const int*)d_in[3];
    const int* imgw  = (const int*)d_in[4];

    float* boxes_w  = (float*)d_ws;                              // 16*25200*4 f32
    float* scores_w = boxes_w + (size_t)BATCH * NANCH * 4;       // 16*25200 f32
    int*   labels_w = (int*)(scores_w + (size_t)BATCH * NANCH);  // 16*25200 i32

    const size_t ldsDecode = (size_t)256 * 85 * sizeof(float);   // 87040 B

    // anchors per level across batch are exact multiples of 256
    yolo_decode<<<1200, 256, ldsDecode, stream>>>(p0, boxes_w, scores_w, labels_w,
                                                  80,  8, 0,     0, imgh, imgw);
    yolo_decode<<< 300, 256, ldsDecode, stream>>>(p1, boxes_w, scores_w, labels_w,
                                                  40, 16, 3, 19200, imgh, imgw);
    yolo_decode<<<  75, 256, ldsDecode, stream>>>(p2, boxes_w, scores_w, labels_w,
                                                  20, 32, 6, 24000, imgh, imgw);

    // 256 KB dynamic LDS: whole padded key array per image in one WGP's LDS
    yolo_nms<<<BATCH, 1024, (size_t)SORT_N * sizeof(unsigned long long), stream>>>(
        boxes_w, scores_w, labels_w, (float*)d_out);
}